// MultiProxProblem_30468497998478
// MI455X (gfx1250) — compile-verified
//
#include <hip/hip_runtime.h>

// Batched LASSO proximal gradient on MI455X (gfx1250), fp32 WMMA path.
//
//   R = X * A^T - B          (N x M)   [GEMM1, A row-access: contiguous K]
//   G = R * A                (N x D)   [GEMM2, uses pre-transposed AT: contiguous K]
//   X = prox_l1(X - ss*G, ss*LAM)
// Output: loss[n] = 0.5 * sum_m R[n,m]^2 with R from final X.
//
// Each wave computes a 32x32 output region (2x2 tiles of 16x16) so every
// operand fragment (one global_load_b64) feeds two v_wmma_f32_16x16x4_f32,
// halving L2 traffic vs one-tile-per-wave. All data is L2-resident (8 MB
// working set vs 192 MB L2), so L2 bandwidth is the binding resource.

#define N_PROB   512
#define M_OBS    1024
#define D_VARS   1024
#define MAX_IT   20
#define LAM_C    0.1f

typedef float v2f __attribute__((ext_vector_type(2)));
typedef float v8f __attribute__((ext_vector_type(8)));

// f32 WMMA fragment layout (wave32):
//  A-operand 16x4 : lane L<16 -> row L, K={0,1} in {v0,v1}; lane L+16 -> row L, K={2,3}
//  B-operand 4x16 : lane L<16 -> col L, K={0,1};            lane L+16 -> col L, K={2,3}
//  C/D 16x16      : VGPR v: lanes 0-15 -> (M=v, N=lane), lanes 16-31 -> (M=v+8, N=lane-16)
__device__ __forceinline__ v8f wmma_f32(v2f a, v2f b, v8f c) {
    return __builtin_amdgcn_wmma_f32_16x16x4_f32(
        false, a, false, b, (short)0, c, false, false);
}

// Tiled transpose: AT[d*M + m] = A[m*D + d].  grid (D/32, M/32), block (32,8).
__global__ void transpose_kernel(const float* __restrict__ A,
                                 float* __restrict__ AT) {
    __shared__ float tile[32][33];
    const int x0 = blockIdx.x * 32;   // d
    const int y0 = blockIdx.y * 32;   // m
    const int tx = threadIdx.x, ty = threadIdx.y;
#pragma unroll
    for (int j = 0; j < 32; j += 8)
        tile[ty + j][tx] = A[(size_t)(y0 + ty + j) * D_VARS + x0 + tx];
    __syncthreads();
#pragma unroll
    for (int j = 0; j < 32; j += 8)
        AT[(size_t)(x0 + ty + j) * M_OBS + y0 + tx] = tile[tx][ty + j];
}

// R[n,m] = sum_d X[n,d]*A[m,d] - B[n,m]
// grid: (M_OBS/128, N_PROB/64), block: 256 (8 waves in 2(n) x 4(m); 32x32/wave)
__global__ void lasso_residual_kernel(const float* __restrict__ X,
                                      const float* __restrict__ A,
                                      const float* __restrict__ B,
                                      float* __restrict__ R) {
    const int wave = threadIdx.x >> 5;
    const int lane = threadIdx.x & 31;
    const int ln   = lane & 15;
    const int koff = (lane >> 4) << 1;                 // 0 or 2
    const int n0   = blockIdx.y * 64 + (wave >> 2) * 32;
    const int m0   = blockIdx.x * 128 + (wave & 3) * 32;

    const float* xr0 = X + (size_t)(n0 + ln) * D_VARS + koff;
    const float* xr1 = xr0 + (size_t)16 * D_VARS;
    const float* ar0 = A + (size_t)(m0 + ln) * D_VARS + koff;
    const float* ar1 = ar0 + (size_t)16 * D_VARS;

    v8f c00 = {}, c01 = {}, c10 = {}, c11 = {};
#pragma unroll 4
    for (int k = 0; k < D_VARS; k += 4) {
        v2f x0 = *(const v2f*)(xr0 + k);
        v2f x1 = *(const v2f*)(xr1 + k);
        v2f a0 = *(const v2f*)(ar0 + k);
        v2f a1 = *(const v2f*)(ar1 + k);
        c00 = wmma_f32(x0, a0, c00);
        c01 = wmma_f32(x0, a1, c01);
        c10 = wmma_f32(x1, a0, c10);
        c11 = wmma_f32(x1, a1, c11);
    }

    const int rowAdd = (lane >> 4) << 3;               // 0 or 8
    const v8f* cc[2][2] = { { &c00, &c01 }, { &c10, &c11 } };
#pragma unroll
    for (int ti = 0; ti < 2; ++ti) {
#pragma unroll
        for (int tj = 0; tj < 2; ++tj) {
            const v8f& c = *cc[ti][tj];
#pragma unroll
            for (int v = 0; v < 8; ++v) {
                const int n = n0 + 16 * ti + v + rowAdd;
                const size_t idx = (size_t)n * M_OBS + m0 + 16 * tj + ln;
                R[idx] = c[v] - B[idx];
            }
        }
    }
}

// G = R*A (via AT rows); X[n,d] = prox_l1(X[n,d] - ss[n]*G[n,d], ss[n]*LAM)
// grid: (D_VARS/128, N_PROB/64), block: 256
__global__ void lasso_grad_update_kernel(const float* __restrict__ R,
                                         const float* __restrict__ AT,
                                         const float* __restrict__ step,
                                         float* __restrict__ X) {
    const int wave = threadIdx.x >> 5;
    const int lane = threadIdx.x & 31;
    const int ln   = lane & 15;
    const int koff = (lane >> 4) << 1;
    const int n0   = blockIdx.y * 64 + (wave >> 2) * 32;
    const int d0   = blockIdx.x * 128 + (wave & 3) * 32;

    const float* rr0 = R + (size_t)(n0 + ln) * M_OBS + koff;
    const float* rr1 = rr0 + (size_t)16 * M_OBS;
    const float* tr0 = AT + (size_t)(d0 + ln) * M_OBS + koff;   // AT[d][k=m]
    const float* tr1 = tr0 + (size_t)16 * M_OBS;

    v8f c00 = {}, c01 = {}, c10 = {}, c11 = {};
#pragma unroll 4
    for (int k = 0; k < M_OBS; k += 4) {
        v2f r0 = *(const v2f*)(rr0 + k);
        v2f r1 = *(const v2f*)(rr1 + k);
        v2f a0 = *(const v2f*)(tr0 + k);
        v2f a1 = *(const v2f*)(tr1 + k);
        c00 = wmma_f32(r0, a0, c00);
        c01 = wmma_f32(r0, a1, c01);
        c10 = wmma_f32(r1, a0, c10);
        c11 = wmma_f32(r1, a1, c11);
    }

    const int rowAdd = (lane >> 4) << 3;
    const v8f* cc[2][2] = { { &c00, &c01 }, { &c10, &c11 } };
#pragma unroll
    for (int ti = 0; ti < 2; ++ti) {
#pragma unroll
        for (int tj = 0; tj < 2; ++tj) {
            const v8f& c = *cc[ti][tj];
#pragma unroll
            for (int v = 0; v < 8; ++v) {
                const int n = n0 + 16 * ti + v + rowAdd;
                const float ss  = step[n];
                const float thr = ss * LAM_C;
                const size_t idx = (size_t)n * D_VARS + d0 + 16 * tj + ln;
                const float u   = X[idx] - ss * c[v];
                const float mag = fmaxf(fabsf(u) - thr, 0.0f);
                X[idx] = copysignf(mag, u);        // sign(u)*max(|u|-thr,0)
            }
        }
    }
}

// loss[n] = 0.5 * sum_m R[n,m]^2 ; one block per problem
__global__ void lasso_loss_kernel(const float* __restrict__ R,
                                  float* __restrict__ out) {
    __shared__ float sd[256];
    const int n   = blockIdx.x;
    const int tid = threadIdx.x;
    float s = 0.0f;
    for (int m = tid; m < M_OBS; m += 256) {
        const float r = R[(size_t)n * M_OBS + m];
        s += r * r;
    }
    sd[tid] = s;
    __syncthreads();
#pragma unroll
    for (int stride = 128; stride > 0; stride >>= 1) {
        if (tid < stride) sd[tid] += sd[tid + stride];
        __syncthreads();
    }
    if (tid == 0) out[n] = 0.5f * sd[0];
}

extern "C" void kernel_launch(void* const* d_in, const int* in_sizes, int n_in,
                              void* d_out, int out_size, void* d_ws, size_t ws_size,
                              hipStream_t stream) {
    const float* A    = (const float*)d_in[0];   // (M_OBS, D_VARS)
    const float* B    = (const float*)d_in[1];   // (N_PROB, M_OBS)
    const float* x0   = (const float*)d_in[2];   // (N_PROB, D_VARS)
    const float* step = (const float*)d_in[3];   // (N_PROB,)
    // d_in[4] = max_iter scalar; fixed at 20 by the problem definition.
    (void)in_sizes; (void)n_in; (void)out_size; (void)ws_size;

    float* Rws  = (float*)d_ws;                        // 512*1024 floats (2 MB)
    float* Xws  = Rws + (size_t)N_PROB * M_OBS;        // 512*1024 floats (2 MB)
    float* ATws = Xws + (size_t)N_PROB * D_VARS;       // 1024*1024 floats (4 MB)

    // One-time per launch: working copy of x0, and A^T for GEMM2's B-operand.
    hipMemcpyAsync(Xws, x0, (size_t)N_PROB * D_VARS * sizeof(float),
                   hipMemcpyDeviceToDevice, stream);
    transpose_kernel<<<dim3(D_VARS / 32, M_OBS / 32), dim3(32, 8), 0, stream>>>(A, ATws);

    const dim3 blk(256);
    const dim3 grdR(M_OBS / 128, N_PROB / 64);    // 8 x 8
    const dim3 grdG(D_VARS / 128, N_PROB / 64);   // 8 x 8

    for (int it = 0; it < MAX_IT; ++it) {
        lasso_residual_kernel<<<grdR, blk, 0, stream>>>(Xws, A, B, Rws);
        lasso_grad_update_kernel<<<grdG, blk, 0, stream>>>(Rws, ATws, step, Xws);
    }
    // Final residual with converged X, then per-problem loss.
    lasso_residual_kernel<<<grdR, blk, 0, stream>>>(Xws, A, B, Rws);
    lasso_loss_kernel<<<N_PROB, 256, 0, stream>>>(Rws, (float*)d_out);
}